// ToepliztV3_15221364097802
// MI455X (gfx1250) — compile-verified
//
#include <hip/hip_runtime.h>

// out[b,h,t,d] = e * sum_{s<=t} x[b,h,s,d]   (the e^-60 tail term is ~1e-27 abs, below f32 ulp)
//
// Scan on the matrix units: 16-row inclusive cumsum tile = (c1*L) * X + splat(carry),
// L = 16x16 lower-triangular ones, done as 4 chained V_WMMA_F32_16X16X4_F32 (K=4 each).
// The scale c1=exp(zero[0]) is folded into L so the WMMA output is the final result and
// the carry recurrence (row M=15 of D) stays self-consistent.

typedef __attribute__((ext_vector_type(2))) float v2f;
typedef __attribute__((ext_vector_type(8))) float v8f;

#define NROWS 4096
#define DCOLS 64

__global__ __launch_bounds__(32) void toeplitz_scan_wmma(
    const float* __restrict__ x, const float* __restrict__ zero,
    float* __restrict__ out)
{
    const int bh   = blockIdx.x >> 2;        // 0..63  : (batch*16 + head)
    const int ct   = blockIdx.x & 3;         // 0..3   : 16-column tile
    const int lane = threadIdx.x & 31;       // wave32
    const int col  = lane & 15;
    const bool hi  = lane >= 16;

    const float c1 = __expf(fminf(fmaxf(zero[0], -60.f), 30.f));

    const size_t base = (size_t)bh * NROWS * DCOLS + (size_t)ct * 16 + (size_t)col;
    const float* xp = x   + base;
    float*       op = out + base;

    // A = c1 * lower-triangular ones, split into 4 K-chunks of 4.
    // 16x4 f32 A layout: lanes 0-15 hold M=0..15, VGPR0 = {K=0 | K=2}, VGPR1 = {K=1 | K=3}.
    v2f a[4];
#pragma unroll
    for (int j = 0; j < 4; ++j) {
        const int kb = 4 * j + (hi ? 2 : 0);
        a[j].x = (col >= kb)     ? c1 : 0.f;
        a[j].y = (col >= kb + 1) ? c1 : 0.f;
    }

    float carry = 0.f;   // running scaled column sum; lane holds carry for its col (both halves)

    for (int r = 0; r < NROWS / 16; ++r) {
        const float* tp = xp + (size_t)r * 16 * DCOLS;

        // Prefetch two row-tiles ahead into near caches (global_prefetch_b8).
        __builtin_prefetch(tp + 2 * 16 * DCOLS, 0, 3);

        // B fragments: 4x16 rows of the tile per K-chunk.
        // Mirrors A layout: lanes 0-15 carry rows {4j, 4j+1}, lanes 16-31 rows {4j+2, 4j+3}.
        v2f b[4];
#pragma unroll
        for (int j = 0; j < 4; ++j) {
            const int row = 4 * j + (hi ? 2 : 0);
            b[j].x = tp[(size_t)row       * DCOLS];
            b[j].y = tp[(size_t)(row + 1) * DCOLS];
        }

        // C = carry broadcast down each column (C/D layout: lane -> column, VGPR -> row).
        v8f acc;
#pragma unroll
        for (int v = 0; v < 8; ++v) acc[v] = carry;

        // D = c1*(L * X) + C  via 4 chained K=4 WMMAs.
#pragma unroll
        for (int j = 0; j < 4; ++j)
            acc = __builtin_amdgcn_wmma_f32_16x16x4_f32(
                false, a[j], false, b[j], (short)0, acc, false, false);

        // Store: scale already folded in. D rows: lanes 0-15 -> M=v, lanes 16-31 -> M=v+8.
        float* topp = op + (size_t)(r * 16 + (hi ? 8 : 0)) * DCOLS;
#pragma unroll
        for (int v = 0; v < 8; ++v)
            topp[(size_t)v * DCOLS] = acc[v];

        // Next carry = row M=15 of D = acc[7] on lanes 16..31 (lane-16 == column).
        // Half-wave swap via v_permlanex16 (pure VALU, keeps LDS pipe off the serial chain):
        // identity selects -> lane i reads lane i^16.
        const int ai = __float_as_int(acc[7]);
        const int sw = __builtin_amdgcn_permlanex16(ai, ai, 0x76543210u, 0xfedcba98u,
                                                    true, false);
        carry = hi ? acc[7] : __int_as_float(sw);
    }
}

extern "C" void kernel_launch(void* const* d_in, const int* in_sizes, int n_in,
                              void* d_out, int out_size, void* d_ws, size_t ws_size,
                              hipStream_t stream) {
    const float* x    = (const float*)d_in[0];  // (4,16,4096,64) f32
    const float* zero = (const float*)d_in[2];  // scalar (value 1.0 in reference setup)
    float* out = (float*)d_out;                 // (4,16,4096,64) f32

    toeplitz_scan_wmma<<<dim3(256), dim3(32), 0, stream>>>(x, zero, out);
}